// SigLayer_32693291057969
// MI455X (gfx1250) — compile-verified
//
#include <hip/hip_runtime.h>

// CDNA5 / gfx1250, wave32. One wave per batch element; depth-4 signature scan.
// Level-4 (64x64) accumulator lives in 16 WMMA f32 16x16 tiles (128 VGPRs),
// updated by one v_wmma_f32_16x16x4_f32 per tile per step (effective K=3:
// rank-1 h(x)e2 + the two S3 columns active in that column tile).

typedef float v2f __attribute__((ext_vector_type(2)));
typedef float v8f __attribute__((ext_vector_type(8)));

__device__ __forceinline__ float bperm(float v, int srcLane) {
  // wave32 lane gather via ds_bpermute (byte index)
  return __int_as_float(__builtin_amdgcn_ds_bpermute(srcLane << 2, __float_as_int(v)));
}

__global__ __launch_bounds__(32) void sig_depth4_kernel(const float* __restrict__ x,
                                                        float* __restrict__ out,
                                                        int L) {
  const int lane = threadIdx.x;          // 0..31
  const int b    = blockIdx.x;
  const float* __restrict__ xp = x + (size_t)b * (size_t)L * 8;

  const bool lo16 = (lane < 16);
  const int  j8   = lane & 7;            // j index (replicated-by-8 layout)
  const int  i4   = lane >> 3;           // i for p=lane (lo half)

  // ---- signature state (all in VGPRs) ----
  // s1[lane&7] replicated; s2 split lo/hi halves of p=0..63.
  float s1 = 0.f, s2_lo = 0.f, s2_hi = 0.f;
  // S3s[r][c]: lanes0-15 hold S3[16r+m][2c], lanes16-31 hold S3[16r+m][2c+1]
  float S3s[4][4];
  // acc[r][c]: WMMA f32 16x16 C/D tile of M4 rows 16r.., cols 16c..
  v8f acc[4][4];
#pragma unroll
  for (int r = 0; r < 4; ++r)
#pragma unroll
    for (int c = 0; c < 4; ++c) {
      S3s[r][c] = 0.f;
      acc[r][c] = (v8f){0.f, 0.f, 0.f, 0.f, 0.f, 0.f, 0.f, 0.f};
    }

  float prev = xp[j8];                   // row 0, replicated by 8

  for (int t = 0; t < L - 1; ++t) {
    const float cur = xp[(size_t)(t + 1) * 8 + j8];
    const float d   = cur - prev;        // increment, lane holds d[lane&7]
    prev = cur;

    // broadcasts for p-indexed quantities (p = lane / lane+32, i = p>>3, j = p&7)
    const float di_lo = bperm(d, i4);        // d[i], lo half
    const float di_hi = bperm(d, i4 + 4);    // d[i], hi half
    const float s1_lo = bperm(s1, i4);       // s1[i] (old)
    const float s1_hi = bperm(s1, i4 + 4);

    const float e2_lo = di_lo * d * 0.5f;    // e2[p] = d_i d_j / 2
    const float e2_hi = di_hi * d * 0.5f;
    const float w_lo  = s1_lo * d;           // w[p] = s1[i] d_j
    const float w_hi  = s1_hi * d;

    const float g_lo = e2_lo * (1.f / 3.f) + w_lo * 0.5f + s2_lo;        // S3 rank-1 left factor
    const float g_hi = e2_hi * (1.f / 3.f) + w_hi * 0.5f + s2_hi;
    const float h_lo = e2_lo * (1.f / 6.f) + w_lo * (1.f / 3.f) + s2_lo; // M4 rank-1 left factor
    const float h_hi = e2_hi * (1.f / 6.f) + w_hi * (1.f / 3.f) + s2_hi;

    // 16-lane half swaps (lane ^ 16)
    const int xl = lane ^ 16;
    const float h_ls  = bperm(h_lo, xl),  h_hs  = bperm(h_hi, xl);
    const float e2_ls = bperm(e2_lo, xl), e2_hs = bperm(e2_hi, xl);
    const float g_ls  = bperm(g_lo, xl),  g_hs  = bperm(g_hi, xl);

    const float hr[4]  = {h_lo, h_ls, h_hi, h_hs};       // h[16r+m] valid in lanes 0-15
    const float e2c[4] = {e2_lo, e2_ls, e2_hi, e2_hs};   // e2[16c+n] valid in lanes 0-15
    const float grep[4] = {                              // g[16r + (lane&15)] in all lanes
        lo16 ? g_lo : g_ls, lo16 ? g_ls : g_lo,
        lo16 ? g_hi : g_hs, lo16 ? g_hs : g_hi};

    // B operand pieces shared across tiles:
    //   B[1][n] = n<8 ? d[n] : 0  -> lanes0-7 hold d, rest 0 (also zeroes K=3 row)
    const float bv1 = (lane < 8) ? d : 0.f;
    //   B[2][n] = n>=8 ? d[n-8] : 0 -> lanes24-31 hold d[lane-24]
    const float dtail = (lane >= 24) ? d : 0.f;

    // ---- level-4: 16 WMMAs, D = A(16x4) * B(4x16) + C, effective K=3 ----
    // Uses OLD S3 (Chen step reads start-of-step state).
#pragma unroll
    for (int c = 0; c < 4; ++c) {
      v2f bop;
      bop[0] = lo16 ? e2c[c] : dtail;   // rows K=0 (e2) / K=2 (d tail)
      bop[1] = bv1;                     // rows K=1 (d head) / K=3 (zero)
#pragma unroll
      for (int r = 0; r < 4; ++r) {
        v2f aop;
        aop[0] = lo16 ? hr[r]     : S3s[r][c]; // K=0: h ; K=2: S3[:,2c+1]
        aop[1] = lo16 ? S3s[r][c] : 0.f;       // K=1: S3[:,2c] ; K=3: 0
        acc[r][c] = __builtin_amdgcn_wmma_f32_16x16x4_f32(
            false, aop, false, bop, (short)0, acc[r][c], false, false);
      }
    }

    // ---- level-3: S3 += g (x) d ----
#pragma unroll
    for (int c = 0; c < 4; ++c) {
      const float dpair = bperm(d, 2 * c + (lo16 ? 0 : 1)); // d[2c] / d[2c+1]
#pragma unroll
      for (int r = 0; r < 4; ++r) S3s[r][c] += grep[r] * dpair;
    }

    // ---- levels 1,2 commit ----
    s2_lo += e2_lo + w_lo;
    s2_hi += e2_hi + w_hi;
    s1 += d;
  }

  // ---- write output: [s1(8), s2(64), s3(512), s4(4096)] per batch ----
  float* __restrict__ ob = out + (size_t)b * 4680;
  if (lane < 8) ob[lane] = s1;
  ob[8 + lane]      = s2_lo;
  ob[8 + 32 + lane] = s2_hi;
#pragma unroll
  for (int r = 0; r < 4; ++r)
#pragma unroll
    for (int c = 0; c < 4; ++c) {
      const int p = 16 * r + (lane & 15);
      const int k = 2 * c + (lane >> 4);
      ob[72 + p * 8 + k] = S3s[r][c];
    }
#pragma unroll
  for (int r = 0; r < 4; ++r)
#pragma unroll
    for (int c = 0; c < 4; ++c) {
      const v8f v = acc[r][c];
#pragma unroll
      for (int vv = 0; vv < 8; ++vv) {
        const int M = lo16 ? vv : (vv + 8);   // C/D tile layout: VGPR v -> M=v / M=v+8
        const int N = lane & 15;
        ob[584 + (16 * r + M) * 64 + 16 * c + N] = v[vv];
      }
    }
}

extern "C" void kernel_launch(void* const* d_in, const int* in_sizes, int n_in,
                              void* d_out, int out_size, void* d_ws, size_t ws_size,
                              hipStream_t stream) {
  (void)n_in; (void)d_ws; (void)ws_size; (void)out_size;
  const float* x = (const float*)d_in[0];
  float* out = (float*)d_out;
  const int L = 512;
  const int B = in_sizes[0] / (L * 8);   // 128
  sig_depth4_kernel<<<B, 32, 0, stream>>>(x, out, L);
}